// Attention_38500086841978
// MI455X (gfx1250) — compile-verified
//
#include <hip/hip_runtime.h>
#include <hip/hip_bf16.h>

// Problem constants (match the JAX reference)
constexpr int DIM   = 512;
constexpr int BATCH = 1024;

typedef __attribute__((ext_vector_type(2))) float v2f;
typedef __attribute__((ext_vector_type(8))) float v8f;

// -------------------------------------------------------------------------
// Kernel 1: Out[m, n] = sum_d X[m, d] * W[n, d] + bias[n]   (i.e. X @ W^T + b)
// One wave (32 threads) computes a 16(M) x 64(N) strip using
// V_WMMA_F32_16X16X4_F32, looping K in steps of 4 (128 WMMAs per acc).
//
// f32 WMMA fragment layouts (CDNA5 ISA 7.12.2):
//   A 16x4 : lane L -> row M=L%16, K = (L/16)*2 + {0,1}   (contiguous pair)
//   B 4x16 : lane L -> col N=L%16, K = (L/16)*2 + {0,1}   (contiguous pair)
//   C 16x16: VGPR r -> M = r + (L/16)*8, N = L%16
// Since B[k][n] = W[n][k], both fragments are contiguous b64 loads.
// -------------------------------------------------------------------------
__global__ __launch_bounds__(32)
void qkv_gemm_wmma(const float* __restrict__ X, const float* __restrict__ W,
                   const float* __restrict__ bias, float* __restrict__ Out)
{
    const int lane  = threadIdx.x;          // 0..31, EXEC all ones (no divergence)
    const int m0    = blockIdx.x * 16;      // 1024/16 = 64 tiles
    const int n0    = blockIdx.y * 64;      // 512/64  = 8 strips
    const int r16   = lane & 15;
    const int khalf = (lane >> 4) << 1;     // 0 or 2

    const float* ap = X + (size_t)(m0 + r16) * DIM + khalf;
    const float* bp[4];
#pragma unroll
    for (int t = 0; t < 4; ++t)
        bp[t] = W + (size_t)(n0 + 16 * t + r16) * DIM + khalf;

    v8f acc[4] = {};

#pragma unroll 4
    for (int k = 0; k < DIM; k += 4) {
        v2f a = *(const v2f*)(ap + k);
#pragma unroll
        for (int t = 0; t < 4; ++t) {
            v2f b = *(const v2f*)(bp[t] + k);
            // 8 args: (neg_a, A, neg_b, B, c_mod, C, reuse_a, reuse_b)
            acc[t] = __builtin_amdgcn_wmma_f32_16x16x4_f32(
                false, a, false, b, (short)0, acc[t], false, false);
        }
    }

    // Epilogue: add bias, scatter per C/D layout.
    const int mbase = m0 + ((lane >> 4) << 3);
#pragma unroll
    for (int t = 0; t < 4; ++t) {
        const int ncol = n0 + 16 * t + r16;
        const float bv = bias[ncol];
        float* op = Out + (size_t)mbase * DIM + ncol;
#pragma unroll
        for (int r = 0; r < 8; ++r)
            op[(size_t)r * DIM] = acc[t][r] + bv;
    }
}

// -------------------------------------------------------------------------
// Kernel 2: rank-1 softmax-attention, fused.
//   out[b,i] = sum_j exp(q_i*k_j/sqrt(D)) * v_j / sum_j exp(q_i*k_j/sqrt(D))
// Scores are bounded (|q_i*k_j|/sqrt(512) <~ 1), so no max-subtraction is
// needed for stability (mathematically identical softmax).
// One block per batch row; k,v interleaved as float2 in LDS (broadcast
// reads); each thread owns rows i=tid and i=tid+256 so one LDS read feeds
// two exp accumulations. 268M v_exp_f32 total -> this is the hot loop.
// -------------------------------------------------------------------------
__global__ __launch_bounds__(256)
void softmax_av(const float* __restrict__ Q, const float* __restrict__ K,
                const float* __restrict__ V, float* __restrict__ Out)
{
    __shared__ float2 kv[DIM];

    const int b   = blockIdx.x;
    const int tid = threadIdx.x;

    const float* kr = K + (size_t)b * DIM;
    const float* vr = V + (size_t)b * DIM;
    for (int j = tid; j < DIM; j += 256)
        kv[j] = make_float2(kr[j], vr[j]);
    __syncthreads();

    constexpr float inv_scale = 0.04419417382415922f; // 1/sqrt(512)
    const float* qr = Q + (size_t)b * DIM;

    const float q0 = qr[tid]       * inv_scale;
    const float q1 = qr[tid + 256] * inv_scale;

    float num0 = 0.f, den0 = 0.f, num1 = 0.f, den1 = 0.f;
#pragma unroll 4
    for (int j = 0; j < DIM; ++j) {
        const float2 p = kv[j];
        const float e0 = __expf(q0 * p.x);
        const float e1 = __expf(q1 * p.x);
        num0 = fmaf(e0, p.y, num0);
        den0 += e0;
        num1 = fmaf(e1, p.y, num1);
        den1 += e1;
    }

    float* orow = Out + (size_t)b * DIM;
    orow[tid]       = num0 / den0;
    orow[tid + 256] = num1 / den1;
}

// -------------------------------------------------------------------------
extern "C" void kernel_launch(void* const* d_in, const int* in_sizes, int n_in,
                              void* d_out, int out_size, void* d_ws, size_t ws_size,
                              hipStream_t stream)
{
    const float* x  = (const float*)d_in[0];
    const float* Wq = (const float*)d_in[1];
    const float* bq = (const float*)d_in[2];
    const float* Wk = (const float*)d_in[3];
    const float* bk = (const float*)d_in[4];
    const float* Wv = (const float*)d_in[5];
    const float* bv = (const float*)d_in[6];
    float* out = (float*)d_out;

    float* Q = (float*)d_ws;                 // 1024*512 f32 = 2 MB
    float* K = Q + (size_t)BATCH * DIM;      // +2 MB
    float* V = K + (size_t)BATCH * DIM;      // +2 MB  (total 6 MB of ws)

    dim3 gGemm(BATCH / 16, DIM / 64);        // 64 x 8 waves
    qkv_gemm_wmma<<<gGemm, 32, 0, stream>>>(x, Wq, bq, Q);
    qkv_gemm_wmma<<<gGemm, 32, 0, stream>>>(x, Wk, bk, K);
    qkv_gemm_wmma<<<gGemm, 32, 0, stream>>>(x, Wv, bv, V);

    softmax_av<<<BATCH, 256, 0, stream>>>(Q, K, V, out);
}